// RGCNEncoder_33758442947200
// MI455X (gfx1250) — compile-verified
//
#include <hip/hip_runtime.h>

// RGCN (num_blocks mode) on gfx1250.
// N=50000, E=800000, D=128, R=8, B=2 (two 64x64 blocks per relation).
// Workspace layout (floats): cnt[N*8] | sums[N*128] | h[N*128]  (~52.8 MB total)

typedef __attribute__((ext_vector_type(2))) float v2f;
typedef __attribute__((ext_vector_type(8))) float v8f;

#define DD 128
#define RR 8
#define APITCH 132  // LDS row pitch (floats): bank-conflict-free wave32 b64 reads

// ---------------------------------------------------------------- utilities
__global__ void zero_f4(float4* __restrict__ p, int n4) {
  int i = blockIdx.x * blockDim.x + threadIdx.x;
  if (i < n4) p[i] = make_float4(0.f, 0.f, 0.f, 0.f);
}

__global__ void relu_k(float* __restrict__ p, int n) {
  int i = blockIdx.x * blockDim.x + threadIdx.x;
  if (i < n) p[i] = fmaxf(p[i], 0.0f);
}

// Per-(dst,relation) edge counts (graph topology only; reused by both layers).
__global__ void count_edges(const int* __restrict__ dst, const int* __restrict__ et,
                            float* __restrict__ cnt, int E) {
  int e = blockIdx.x * blockDim.x + threadIdx.x;
  if (e < E) atomicAdd(&cnt[dst[e] * RR + et[e]], 1.0f);
}

// Scatter-add source features of edges with etype==rel into sums[dst].
// One wave (32 lanes x float4 = 128 floats) per edge -> coalesced gather,
// whole wave early-outs on relation mismatch.
__global__ void scatter_rel(const float* __restrict__ xin, const int* __restrict__ src,
                            const int* __restrict__ dst, const int* __restrict__ et,
                            float* __restrict__ sums, int E, int rel) {
  int tid = blockIdx.x * blockDim.x + threadIdx.x;
  int e = tid >> 5;
  int c = (tid & 31) * 4;
  if (e >= E) return;
  if (et[e] != rel) return;
  const float4 v = *(const float4*)(xin + (size_t)src[e] * DD + c);
  float* o = sums + (size_t)dst[e] * DD + c;
  atomicAdd(o + 0, v.x);
  atomicAdd(o + 1, v.y);
  atomicAdd(o + 2, v.z);
  atomicAdd(o + 3, v.w);
}

// out[N x 128] = A[N x 128] @ Wr[128 x 128] + bias, via V_WMMA_F32_16X16X4_F32.
// blockDim = (32, 8): one wave per 16x16 output tile, threadIdx.y = column tile.
// The 16x128 A-tile is shared by all 8 waves -> stage it into LDS once.
// f32 WMMA layouts (ISA 7.12.2): A 16x4 -> lane<16 holds K=k0,k0+1; lane>=16 K=k0+2,k0+3.
// C/D: col = lane&15, row = vgpr_index + 8*(lane>>4).
__global__ void gemm_root_wmma(const float* __restrict__ A, const float* __restrict__ Wr,
                               const float* __restrict__ bias, float* __restrict__ out,
                               int N) {
  __shared__ float As[16 * APITCH];
  const int row0 = blockIdx.x * 16;
  // cooperative stage: 256 threads, thread t copies 8 floats of row t/16
  {
    const int t   = threadIdx.y * 32 + threadIdx.x;
    const int row = t >> 4;
    const int c0  = (t & 15) * 8;
    const float4* g = (const float4*)(A + (size_t)(row0 + row) * DD + c0);
    float4 v0 = g[0], v1 = g[1];
    float* s = As + row * APITCH + c0;
    *(float4*)(s)     = v0;
    *(float4*)(s + 4) = v1;
  }
  __syncthreads();

  const int lane = threadIdx.x;          // 0..31
  const int m    = lane & 15;
  const int half = lane >> 4;
  const int col0 = threadIdx.y * 16;     // 8 tiles cover 128 cols
  const float* ar = As + m * APITCH + 2 * half;
  v8f acc = {};
  for (int k0 = 0; k0 < DD; k0 += 4) {
    v2f a; a.x = ar[k0]; a.y = ar[k0 + 1];
    v2f b; b.x = Wr[(k0 + 2 * half) * DD + col0 + m];
           b.y = Wr[(k0 + 2 * half + 1) * DD + col0 + m];
    acc = __builtin_amdgcn_wmma_f32_16x16x4_f32(false, a, false, b, (short)0, acc,
                                                false, false);
  }
  const float bv = bias[col0 + m];
#pragma unroll
  for (int i = 0; i < 8; ++i) {
    int row = row0 + i + 8 * half;
    out[(size_t)row * DD + col0 + m] = acc[i] + bv;
  }
}

// out[n, blk*64 + j] += mean[n, blk*64 + k] * W[rel,blk,k,j],
// mean = sums / max(cnt,1) computed once during the LDS staging pass.
// One wave per 16x16 tile; threadIdx.y in 0..7 encodes (block, column-tile).
__global__ void transform_rel(const float* __restrict__ sums, const float* __restrict__ cnt,
                              const float* __restrict__ W, float* __restrict__ out,
                              int N, int rel) {
  __shared__ float As[16 * APITCH];
  const int row0 = blockIdx.x * 16;
  // stage mean tile (division folded in here, once instead of 8x)
  {
    const int t   = threadIdx.y * 32 + threadIdx.x;
    const int row = t >> 4;
    const int c0  = (t & 15) * 8;
    const float inv = 1.0f / fmaxf(cnt[(size_t)(row0 + row) * RR + rel], 1.0f);
    const float4* g = (const float4*)(sums + (size_t)(row0 + row) * DD + c0);
    float4 v0 = g[0], v1 = g[1];
    v0.x *= inv; v0.y *= inv; v0.z *= inv; v0.w *= inv;
    v1.x *= inv; v1.y *= inv; v1.z *= inv; v1.w *= inv;
    float* s = As + row * APITCH + c0;
    *(float4*)(s)     = v0;
    *(float4*)(s + 4) = v1;
  }
  __syncthreads();

  const int lane = threadIdx.x;
  const int m    = lane & 15;
  const int half = lane >> 4;
  const int t    = threadIdx.y;          // 0..7
  const int blk  = t >> 2;               // which 64x64 block
  const int col0 = (t & 3) * 16;         // column tile inside block
  const float* ar = As + m * APITCH + blk * 64 + 2 * half;
  const float* wp = W + (size_t)((rel * 2 + blk) * 64) * 64;
  v8f acc = {};
  for (int k0 = 0; k0 < 64; k0 += 4) {
    v2f a; a.x = ar[k0]; a.y = ar[k0 + 1];
    v2f b; b.x = wp[(k0 + 2 * half) * 64 + col0 + m];
           b.y = wp[(k0 + 2 * half + 1) * 64 + col0 + m];
    acc = __builtin_amdgcn_wmma_f32_16x16x4_f32(false, a, false, b, (short)0, acc,
                                                false, false);
  }
#pragma unroll
  for (int i = 0; i < 8; ++i) {
    int row = row0 + i + 8 * half;
    float* o = out + (size_t)row * DD + blk * 64 + col0 + m;
    *o += acc[i];
  }
}

// ---------------------------------------------------------------- launcher
extern "C" void kernel_launch(void* const* d_in, const int* in_sizes, int n_in,
                              void* d_out, int out_size, void* d_ws, size_t ws_size,
                              hipStream_t stream) {
  const float* x     = (const float*)d_in[0];
  const int*   ei    = (const int*)d_in[1];   // [2, E]
  const int*   et    = (const int*)d_in[2];   // [E]
  const float* W1    = (const float*)d_in[3]; // [R, B, 64, 64]
  const float* root1 = (const float*)d_in[4]; // [128, 128]
  const float* b1    = (const float*)d_in[5];
  const float* W2    = (const float*)d_in[6];
  const float* root2 = (const float*)d_in[7];
  const float* b2    = (const float*)d_in[8];
  float* out = (float*)d_out;

  const int N = in_sizes[0] / DD;   // 50000 (multiple of 16)
  const int E = in_sizes[2];        // 800000
  const int* src = ei;
  const int* dst = ei + E;

  float* cnt  = (float*)d_ws;                  // N*R
  float* sums = cnt + (size_t)N * RR;          // N*D
  float* h    = sums + (size_t)N * DD;         // N*D

  const int rowTiles = N / 16;
  const dim3 gb(32, 8);
  const int nSum4 = (N * DD) / 4;
  const int sumZB = (nSum4 + 255) / 256;
  const int scatB = (int)(((size_t)E * 32 + 255) / 256);

  // ---- per-(node, relation) counts (shared by both layers)
  {
    int n4 = (N * RR) / 4;
    zero_f4<<<(n4 + 255) / 256, 256, 0, stream>>>((float4*)cnt, n4);
    count_edges<<<(E + 255) / 256, 256, 0, stream>>>(dst, et, cnt, E);
  }

  // ---- layer 1:  h = relu( sum_r W1_r * mean_r(x) + x@root1 + b1 )
  gemm_root_wmma<<<rowTiles, gb, 0, stream>>>(x, root1, b1, h, N);
  for (int r = 0; r < RR; ++r) {
    zero_f4<<<sumZB, 256, 0, stream>>>((float4*)sums, nSum4);
    scatter_rel<<<scatB, 256, 0, stream>>>(x, src, dst, et, sums, E, r);
    transform_rel<<<rowTiles, gb, 0, stream>>>(sums, cnt, W1, h, N, r);
  }
  relu_k<<<(N * DD + 255) / 256, 256, 0, stream>>>(h, N * DD);

  // ---- layer 2:  out = sum_r W2_r * mean_r(h) + h@root2 + b2
  gemm_root_wmma<<<rowTiles, gb, 0, stream>>>(h, root2, b2, out, N);
  for (int r = 0; r < RR; ++r) {
    zero_f4<<<sumZB, 256, 0, stream>>>((float4*)sums, nSum4);
    scatter_rel<<<scatB, 256, 0, stream>>>(h, src, dst, et, sums, E, r);
    transform_rel<<<rowTiles, gb, 0, stream>>>(sums, cnt, W2, out, N, r);
  }
}